// MSSSIMLoss_34248069219065
// MI455X (gfx1250) — compile-verified
//
#include <hip/hip_runtime.h>
#include <hip/hip_bf16.h>

// ---------------------------------------------------------------------------
// MS-SSIM loss for 16x3x512x512 fp32 pairs on gfx1250 (MI455X).
// Separable 7-tap Gaussian conv mapped to V_WMMA_F32_16X16X4_F32 (full-fp32
// matrix pipe) via banded constant matrices. wave32 blocks of 4 waves.
// ---------------------------------------------------------------------------

typedef float v2f __attribute__((ext_vector_type(2)));
typedef float v8f __attribute__((ext_vector_type(8)));

// 7-tap Gaussian, sigma=1.5, normalized (precomputed).
__device__ __forceinline__ float gval(int d) {
    float r = 0.f;
    r = (d == 0 || d == 6) ? 0.03663285f : r;
    r = (d == 1 || d == 5) ? 0.11128076f : r;
    r = (d == 2 || d == 4) ? 0.21674532f : r;
    r = (d == 3)           ? 0.27068215f : r;
    return r;
}

// One separable conv (horizontal WMMA x12 -> LDS relayout -> vertical WMMA x6)
// Q selects the plane: 0:x 1:y 2:x*x 3:y*y 4:x*y
template <int Q>
__device__ __forceinline__ v8f convQ(const v2f (&xh)[2][6], const v2f (&yh)[2][6],
                                     const v2f (&bv)[6], float* mid, int hi, int j) {
#pragma unroll
    for (int m = 0; m < 2; ++m) {
        v8f acc;
#pragma unroll
        for (int i = 0; i < 8; ++i) acc[i] = 0.f;
#pragma unroll
        for (int kk = 0; kk < 6; ++kk) {
            v2f xa = xh[m][kk], ya = yh[m][kk], a;
            if      (Q == 0) a = xa;
            else if (Q == 1) a = ya;
            else if (Q == 2) a = xa * xa;
            else if (Q == 3) a = ya * ya;
            else             a = xa * ya;
            acc = __builtin_amdgcn_wmma_f32_16x16x4_f32(
                false, a, false, bv[kk], (short)0, acc, false, false);
        }
#pragma unroll
        for (int v = 0; v < 8; ++v)
            mid[(m * 16 + v + 8 * hi) * 16 + j] = acc[v];
    }
    v8f out;
#pragma unroll
    for (int i = 0; i < 8; ++i) out[i] = 0.f;
#pragma unroll
    for (int kk = 0; kk < 6; ++kk) {
        v2f b;
        b.x = mid[(4 * kk + 0 + 2 * hi) * 16 + j];
        b.y = mid[(4 * kk + 1 + 2 * hi) * 16 + j];
        out = __builtin_amdgcn_wmma_f32_16x16x4_f32(
            false, bv[kk], false, b, (short)0, out, false, false);
    }
    return out;
}

__global__ __launch_bounds__(128) void ssim_level_kernel(
    const float* __restrict__ x, const float* __restrict__ y,
    float* __restrict__ ssim_acc, float* __restrict__ cs_acc,
    int S, int tilesX) {
    __shared__ float mid_s[4][32 * 16];

    const int lane = threadIdx.x & 31;
    const int wave = threadIdx.x >> 5;
    const int V = S - 6;
    const int nT = tilesX * tilesX;
    const int tile = blockIdx.x * 4 + wave;
    if (tile >= nT) return;                 // uniform per-wave exit: EXEC stays all-1s

    const int plane = blockIdx.y;           // b*3 + c
    const int b = plane / 3;
    const int tY = tile / tilesX, tX = tile % tilesX;
    const int r0 = tY * 16, c0 = tX * 16;
    const float* px = x + (size_t)plane * S * S;
    const float* py = y + (size_t)plane * S * S;
    const int hi = lane >> 4, j = lane & 15;

    // Constant band fragments: serve both as horizontal B operand and
    // vertical A operand (Gv[r][c] = Bh[c][r], identical lane mapping).
    v2f bv[6];
#pragma unroll
    for (int kk = 0; kk < 6; ++kk) {
        int c = 4 * kk + 2 * hi;
        bv[kk].x = gval(c - j);
        bv[kk].y = gval(c + 1 - j);
    }

    // Load the 32x24 input patch directly in WMMA A-fragment layout.
    v2f xh[2][6], yh[2][6];
#pragma unroll
    for (int m = 0; m < 2; ++m) {
        int gr = r0 + m * 16 + j;
        gr = gr > S - 1 ? S - 1 : gr;       // clamp: clamped rows only feed masked outputs
#pragma unroll
        for (int kk = 0; kk < 6; ++kk) {
            int gc = c0 + 4 * kk + 2 * hi;
            gc = gc > S - 2 ? S - 2 : gc;   // keep float2 load in-bounds (even, 8B aligned)
            size_t off = (size_t)gr * S + gc;
            xh[m][kk] = *reinterpret_cast<const v2f*>(px + off);
            yh[m][kk] = *reinterpret_cast<const v2f*>(py + off);
        }
    }

    float* mid = &mid_s[wave][0];
    v8f mu1 = convQ<0>(xh, yh, bv, mid, hi, j);
    v8f mu2 = convQ<1>(xh, yh, bv, mid, hi, j);
    v8f sxx = convQ<2>(xh, yh, bv, mid, hi, j);
    v8f syy = convQ<3>(xh, yh, bv, mid, hi, j);
    v8f sxy = convQ<4>(xh, yh, bv, mid, hi, j);

    const float C1v = 1e-4f, C2v = 9e-4f;
    float ss = 0.f, cc = 0.f;
#pragma unroll
    for (int v = 0; v < 8; ++v) {
        int orow = r0 + v + 8 * hi;
        int ocol = c0 + j;
        float m1 = mu1[v], m2 = mu2[v];
        float s1  = sxx[v] - m1 * m1;
        float s2  = syy[v] - m2 * m2;
        float s12 = sxy[v] - m1 * m2;
        float v1 = 2.f * s12 + C2v;
        float v2 = s1 + s2 + C2v;
        float csv = v1 / v2;
        float ssv = ((2.f * m1 * m2 + C1v) * v1) / ((m1 * m1 + m2 * m2 + C1v) * v2);
        bool valid = (orow < V) && (ocol < V);
        ss += valid ? ssv : 0.f;
        cc += valid ? csv : 0.f;
    }
    // wave32 reduction
#pragma unroll
    for (int off = 16; off > 0; off >>= 1) {
        ss += __shfl_down(ss, off, 32);
        cc += __shfl_down(cc, off, 32);
    }
    if (lane == 0) {
        atomicAdd(&ssim_acc[b], ss);
        atomicAdd(&cs_acc[b], cc);
    }
}

__global__ void pool2_kernel(const float* __restrict__ in, float* __restrict__ out,
                             int Sout, int planes) {
    int idx = blockIdx.x * blockDim.x + threadIdx.x;
    int total = planes * Sout * Sout;
    if (idx >= total) return;
    int oc = idx % Sout;
    int t  = idx / Sout;
    int orow = t % Sout;
    int p = t / Sout;
    int Sin = Sout * 2;
    const float* pi = in + (size_t)p * Sin * Sin;
    size_t o = (size_t)(orow * 2) * Sin + (size_t)(oc * 2);
    out[idx] = 0.25f * (pi[o] + pi[o + 1] + pi[o + Sin] + pi[o + Sin + 1]);
}

__global__ void zero_kernel(float* __restrict__ p, int n) {
    int i = blockIdx.x * blockDim.x + threadIdx.x;
    if (i < n) p[i] = 0.f;
}

__global__ void final_kernel(const float* __restrict__ acc, float* __restrict__ out, int B) {
    int b = threadIdx.x;
    float p = 0.f;
    if (b < B) {
        const float w0 = 0.0448f;
        const float wl[4] = {0.2856f, 0.3001f, 0.2363f, 0.1333f};
        int V4 = (512 >> 4) - 6;
        float cnt4 = 3.f * (float)V4 * (float)V4;
        float v = acc[4 * 16 + b] / cnt4;             // ssim at last level
        p = powf(fmaxf(v, 1e-6f), w0);
#pragma unroll
        for (int l = 0; l < 4; ++l) {                  // cs at levels 0..3
            int Vl = (512 >> l) - 6;
            float cnt = 3.f * (float)Vl * (float)Vl;
            float vv = acc[80 + l * 16 + b] / cnt;
            p *= powf(fmaxf(vv, 1e-6f), wl[l]);
        }
    }
#pragma unroll
    for (int off = 16; off > 0; off >>= 1) p += __shfl_down(p, off, 32);
    if (b == 0) out[0] = 1.f - p / (float)B;
}

extern "C" void kernel_launch(void* const* d_in, const int* in_sizes, int n_in,
                              void* d_out, int out_size, void* d_ws, size_t ws_size,
                              hipStream_t stream) {
    (void)n_in; (void)out_size; (void)ws_size;
    const float* x0 = (const float*)d_in[0];
    const float* y0 = (const float*)d_in[1];
    float* wsf = (float*)d_ws;

    const int B = in_sizes[0] / (3 * 512 * 512);   // 16
    const int planes = B * 3;

    // Workspace layout: [0,80) ssim_acc[5][16], [80,160) cs_acc[5][16],
    // then pyramid x/y buffers for levels 1..4.
    float* acc = wsf;
    size_t off = 160;
    const float* xs[5];
    const float* ys[5];
    float* xw[5];
    float* yw[5];
    xs[0] = x0; ys[0] = y0; xw[0] = nullptr; yw[0] = nullptr;
    for (int l = 1; l < 5; ++l) {
        int S = 512 >> l;
        size_t n = (size_t)planes * S * S;
        xw[l] = wsf + off; off += n;
        yw[l] = wsf + off; off += n;
        xs[l] = xw[l]; ys[l] = yw[l];
    }

    zero_kernel<<<1, 256, 0, stream>>>(acc, 160);

    for (int l = 0; l < 5; ++l) {
        int S = 512 >> l;
        int V = S - 6;
        int tilesX = (V + 15) / 16;
        int nT = tilesX * tilesX;
        dim3 grid((nT + 3) / 4, planes);
        ssim_level_kernel<<<grid, 128, 0, stream>>>(
            xs[l], ys[l], acc + l * 16, acc + 80 + l * 16, S, tilesX);
        if (l < 4) {
            int Sout = S / 2;
            int total = planes * Sout * Sout;
            int blocks = (total + 255) / 256;
            pool2_kernel<<<blocks, 256, 0, stream>>>(xs[l], xw[l + 1], Sout, planes);
            pool2_kernel<<<blocks, 256, 0, stream>>>(ys[l], yw[l + 1], Sout, planes);
        }
    }

    final_kernel<<<1, 32, 0, stream>>>(acc, (float*)d_out, B);
}